// ModelSGLang_68186900792087
// MI455X (gfx1250) — compile-verified
//
#include <hip/hip_runtime.h>

typedef __attribute__((ext_vector_type(2))) float v2f;
typedef __attribute__((ext_vector_type(8))) float v8f;

// Problem dims (match reference)
constexpr int Bdim = 4, Tdim = 2048, Hdim = 8, Kdim = 128, HVdim = 16, Vdim = 128;
constexpr int CH = 16;                   // chunk length (== WMMA tile M)
constexpr int NCHUNK = Tdim / CH;        // 128 sequential chunks
constexpr float QSCALE = 0.08838834764831845f;  // 128^-0.5

// ---- LDS layouts ----
// S and U store K-row PAIRS interleaved: element (k, n) lives at
//   (k>>1)*PAIR_STRIDE + 2*n + (k&1)
// so a WMMA B-fragment {B[k][n], B[k+1][n]} (k even) is one contiguous 8B ds_load_b64.
// PAIR_STRIDE % 64 == 32 -> the two half-wave row-pairs land in disjoint bank halves.
constexpr int SP = 2 * Vdim + 32;        // 288 floats per S row-pair (64 pairs)
constexpr int UP = 2 * Vdim + 32;        // 288 floats per U row-pair (8 pairs)
constexpr int KQ_STRIDE = 132;           // row-major K/Q tiles; %64==4 -> conflict-free A-frags

__device__ __forceinline__ int sidx(int k, int n) { return (k >> 1) * SP + 2 * n + (k & 1); }
__device__ __forceinline__ int uidx(int t, int n) { return (t >> 1) * UP + 2 * n + (t & 1); }

constexpr int SMEM_FLOATS = (Kdim / 2) * SP        // S        18432
                          + CH * KQ_STRIDE         // Kc        2112
                          + CH * KQ_STRIDE         // Qc        2112
                          + (CH / 2) * UP          // U         2304
                          + 256                    // Amat
                          + 256                    // Attn
                          + 4 * CH;                // gcum/gamv/ghat/beta

__device__ __forceinline__ v8f wmma4(v2f a, v2f b, v8f c) {
  // D = A(16x4, f32) * B(4x16, f32) + C(16x16, f32)
  return __builtin_amdgcn_wmma_f32_16x16x4_f32(false, a, false, b, (short)0, c,
                                               false, false);
}

__global__ __launch_bounds__(256)
void gated_delta_chunked(const float* __restrict__ A_log,
                         const float* __restrict__ a_in,
                         const float* __restrict__ dt_bias,
                         const float* __restrict__ q_in,
                         const float* __restrict__ k_in,
                         const float* __restrict__ v_in,
                         const float* __restrict__ b_in,
                         const float* __restrict__ init_src,
                         const long long* __restrict__ init_idx,
                         float* __restrict__ out)
{
  extern __shared__ __align__(16) float smem[];
  float* S    = smem;                         // paired-row state  K x V
  float* Kc   = S    + (Kdim / 2) * SP;       // [16][KQ_STRIDE] chunk K (row major)
  float* Qc   = Kc   + CH * KQ_STRIDE;        // [16][KQ_STRIDE] chunk Q (pre-scaled)
  float* Uc   = Qc   + CH * KQ_STRIDE;        // paired-row rhs -> solved U
  float* Amat = Uc   + (CH / 2) * UP;         // [16][16] strict-lower solve matrix
  float* Attn = Amat + 256;                   // [16][16] causal (incl diag) attn
  float* gcum = Attn + 256;                   // [16] cumulative log-decay
  float* gamv = gcum + CH;                    // [16] exp(gcum[t])      (= Gamma_t)
  float* ghat = gamv + CH;                    // [16] exp(gcum[15]-gcum[t])
  float* betav= ghat + CH;                    // [16] sigmoid(b)

  const int tid  = threadIdx.x;
  const int wave = tid >> 5;
  const int lane = tid & 31;
  const int half = lane >> 4;   // 0: lanes 0-15, 1: lanes 16-31
  const int l16  = lane & 15;

  const int head = blockIdx.x;         // B*HV heads, one workgroup each
  const int b    = head / HVdim;
  const int hv   = head % HVdim;
  const int h    = hv / (HVdim / Hdim);  // GQA: value-head -> qk-head

  // ---------------- initial state ----------------
  const long long sidx64 = init_idx[b];
  for (int i = tid; i < Kdim * Vdim; i += blockDim.x) {
    const int kk = i >> 7, vv = i & 127;
    float val = 0.f;
    if (sidx64 >= 0)
      val = init_src[(((int)sidx64 * HVdim + hv) * Kdim + kk) * Vdim + vv];
    S[sidx(kk, vv)] = val;
  }
  const float negA = -__expf(A_log[hv]);
  const float dtb  = dt_bias[hv];
  __syncthreads();

  for (int c0 = 0; c0 < NCHUNK; ++c0) {
    const int t0 = c0 * CH;

    // ---- gates (wave 0, lane-parallel + shfl prefix-scan) ----
    // g = -exp(A_log)*softplus(a+dt_bias); beta = sigmoid(b)
    if (wave == 0) {
      float g = 0.f, beta = 0.f;
      if (lane < CH) {
        const int bt = (b * Tdim + t0 + lane) * HVdim + hv;
        const float x  = a_in[bt] + dtb;
        const float sp = (x <= 20.f) ? log1pf(__expf(x)) : x;
        g    = negA * sp;
        beta = 1.f / (1.f + __expf(-b_in[bt]));
      }
      #pragma unroll
      for (int off = 1; off < CH; off <<= 1) {      // inclusive scan over lanes 0..15
        const float nb = __shfl_up(g, off, 32);
        if (lane >= off) g += nb;
      }
      const float run = __shfl(g, CH - 1, 32);
      if (lane < CH) {
        gcum[lane]  = g;
        gamv[lane]  = __expf(g);
        ghat[lane]  = __expf(run - g);
        betav[lane] = beta;
      }
    }

    // ---- stage K / Q chunk tiles into LDS; prefetch next chunk ----
    for (int i = tid; i < CH * Kdim; i += blockDim.x) {
      const int r = i >> 7, cc = i & 127;
      const int base = ((b * Tdim + t0 + r) * Hdim + h) * Kdim + cc;
      Kc[r * KQ_STRIDE + cc] = k_in[base];
      Qc[r * KQ_STRIDE + cc] = q_in[base] * QSCALE;
      if (lane == 0 && c0 + 1 < NCHUNK) {           // global_prefetch_b8 for chunk c0+1
        __builtin_prefetch(k_in + base + CH * Hdim * Kdim, 0, 0);
        __builtin_prefetch(q_in + base + CH * Hdim * Kdim, 0, 0);
      }
    }
    __syncthreads();

    // ---- KS0 = K@S0, QS0 = Q@S0 : each wave owns output cols n0..n0+15 ----
    const int n0 = wave * 16;
    v8f ks = {}; v8f qs = {};
    #pragma unroll
    for (int kk = 0; kk < Kdim; kk += 4) {
      const int krow = kk + 2 * half;               // even
      v2f ak, aq;
      ak.x = Kc[l16 * KQ_STRIDE + krow];
      ak.y = Kc[l16 * KQ_STRIDE + krow + 1];
      aq.x = Qc[l16 * KQ_STRIDE + krow];
      aq.y = Qc[l16 * KQ_STRIDE + krow + 1];
      const v2f bs = *(const v2f*)&S[sidx(krow, n0 + l16)];  // one ds_load_b64
      ks = wmma4(ak, bs, ks);
      qs = wmma4(aq, bs, qs);
    }

    // ---- rhs_t = beta_t * (v_t - Gamma_t * (K S0)_t)  -> Uc (paired stores) ----
    #pragma unroll
    for (int i = 0; i < 8; i += 2) {
      const int r   = i + 8 * half;                 // even
      const int col = n0 + l16;
      const int vb  = ((b * Tdim + t0 + r) * HVdim + hv) * Vdim + col;
      const float v0 = v_in[vb];
      const float v1 = v_in[vb + HVdim * Vdim];
      v2f up;
      up.x = betav[r]     * (v0 - gamv[r]     * ks[i]);
      up.y = betav[r + 1] * (v1 - gamv[r + 1] * ks[i + 1]);
      *(v2f*)&Uc[uidx(r, col)] = up;                // one ds_store_b64
      if (l16 == 0 && c0 + 1 < NCHUNK)              // prefetch next chunk's V rows
        __builtin_prefetch(v_in + vb + CH * HVdim * Vdim, 0, 0);
    }

    // ---- KK^T (wave 0) and QK^T (wave 1), masked+scaled into LDS ----
    if (wave < 2) {
      const float* Mc = (wave == 0) ? Kc : Qc;
      v8f acc = {};
      #pragma unroll
      for (int kk = 0; kk < Kdim; kk += 4) {
        const int krow = kk + 2 * half;
        v2f aa, bb;
        aa.x = Mc[l16 * KQ_STRIDE + krow];
        aa.y = Mc[l16 * KQ_STRIDE + krow + 1];
        bb.x = Kc[l16 * KQ_STRIDE + krow];          // B = K^T (transposed LDS read)
        bb.y = Kc[l16 * KQ_STRIDE + krow + 1];
        acc = wmma4(aa, bb, acc);
      }
      #pragma unroll
      for (int i = 0; i < 8; ++i) {
        const int t = i + 8 * half, s = l16;
        if (wave == 0)  // strict lower: A[t][s] = beta_t*(G_t/G_s)*(k_t.k_s)
          Amat[t * 16 + s] = (s < t) ? betav[t] * __expf(gcum[t] - gcum[s]) * acc[i] : 0.f;
        else            // causal incl diag: Attn[t][s] = (G_t/G_s)*(q_t.k_s)
          Attn[t * 16 + s] = (s <= t) ? __expf(gcum[t] - gcum[s]) * acc[i] : 0.f;
      }
    }
    __syncthreads();

    // ---- forward-substitution solve (I + A) u = rhs; per-column, no barriers ----
    if (tid < Vdim) {
      float rhs[CH], u[CH];
      #pragma unroll
      for (int tp = 0; tp < CH; tp += 2) {          // paired b64 loads
        const v2f p = *(const v2f*)&Uc[uidx(tp, tid)];
        rhs[tp] = p.x; rhs[tp + 1] = p.y;
      }
      u[0] = rhs[0];
      #pragma unroll
      for (int t = 1; t < CH; ++t) {
        float acc = rhs[t];
        #pragma unroll
        for (int s = 0; s < CH - 1; ++s)
          if (s < t) acc -= Amat[t * 16 + s] * u[s];
        u[t] = acc;
      }
      #pragma unroll
      for (int tp = 0; tp < CH; tp += 2) {          // paired b64 stores
        v2f p; p.x = u[tp]; p.y = u[tp + 1];
        *(v2f*)&Uc[uidx(tp, tid)] = p;
      }
    }
    __syncthreads();

    // ---- O = diag(Gamma)*QS0 + Attn @ U  (QS0 kept in registers) ----
    v8f oc;
    #pragma unroll
    for (int i = 0; i < 8; ++i)
      oc[i] = gamv[i + 8 * half] * qs[i];
    #pragma unroll
    for (int kk = 0; kk < CH; kk += 4) {
      const int krow = kk + 2 * half;
      v2f aa;
      aa.x = Attn[l16 * 16 + krow];
      aa.y = Attn[l16 * 16 + krow + 1];
      const v2f bb = *(const v2f*)&Uc[uidx(krow, n0 + l16)];
      oc = wmma4(aa, bb, oc);
    }
    #pragma unroll
    for (int i = 0; i < 8; ++i) {
      const int r = i + 8 * half;
      out[((b * Tdim + t0 + r) * HVdim + hv) * Vdim + n0 + l16] = oc[i];
    }

    // ---- state: S = Gamma_C * S + (diag(ghat) K)^T @ U ----
    const float gammaC = gamv[CH - 1];
    const int k0 = wave * 16;             // each wave owns 16 state rows
    #pragma unroll
    for (int ct = 0; ct < 8; ++ct) {      // Kc A-frags invariant across ct -> CSE'd
      const int nn0 = ct * 16;
      v8f sc;
      #pragma unroll
      for (int i = 0; i < 8; i += 2) {    // paired b64 reads of old state
        const v2f sp = *(const v2f*)&S[sidx(k0 + i + 8 * half, nn0 + l16)];
        sc[i] = gammaC * sp.x; sc[i + 1] = gammaC * sp.y;
      }
      #pragma unroll
      for (int kk = 0; kk < CH; kk += 4) {
        const int srow = kk + 2 * half;
        v2f aa;
        aa.x = Kc[srow * KQ_STRIDE + k0 + l16] * ghat[srow];       // A = K^T * ghat
        aa.y = Kc[(srow + 1) * KQ_STRIDE + k0 + l16] * ghat[srow + 1];
        const v2f bb = *(const v2f*)&Uc[uidx(srow, nn0 + l16)];
        sc = wmma4(aa, bb, sc);
      }
      #pragma unroll
      for (int i = 0; i < 8; i += 2) {    // paired b64 state write-back
        v2f sp; sp.x = sc[i]; sp.y = sc[i + 1];
        *(v2f*)&S[sidx(k0 + i + 8 * half, nn0 + l16)] = sp;
      }
    }
    __syncthreads();   // S / Kc / Uc reuse next chunk
  }
}

extern "C" void kernel_launch(void* const* d_in, const int* in_sizes, int n_in,
                              void* d_out, int out_size, void* d_ws, size_t ws_size,
                              hipStream_t stream) {
  const float* A_log   = (const float*)d_in[0];
  const float* a       = (const float*)d_in[1];
  const float* dt_bias = (const float*)d_in[2];
  const float* q       = (const float*)d_in[3];
  const float* k       = (const float*)d_in[4];
  const float* v       = (const float*)d_in[5];
  const float* bgate   = (const float*)d_in[6];
  const float* init    = (const float*)d_in[7];
  const long long* idx = (const long long*)d_in[8];
  float* out = (float*)d_out;

  const size_t smem = (size_t)SMEM_FLOATS * sizeof(float);  // ~100 KB of 320 KB/WGP
  dim3 grid(Bdim * HVdim);   // 64 heads, one workgroup each
  gated_delta_chunked<<<grid, 256, smem, stream>>>(A_log, a, dt_bias, q, k, v,
                                                   bgate, init, idx, out);
}